// SplineConv_27977416966689
// MI455X (gfx1250) — compile-verified
//
#include <hip/hip_runtime.h>

// ---------------------------------------------------------------------------
// SplineConv for MI455X (gfx1250, wave32).
//  - Bucket (node,tap) pairs by weight index -> dense WMMA GEMM per bucket.
//  - fp32 WMMA (v_wmma_f32_16x16x4_f32): exact precision, problem is tiny
//    (~1.7 GFLOP over ~17MB; everything L2-resident at 192MB).
//  - basis*mask folded into the A operand; root/bias/mask folded into the
//    output initialization GEMM.
// ---------------------------------------------------------------------------

typedef float v2f __attribute__((ext_vector_type(2)));
typedef float v8f __attribute__((ext_vector_type(8)));

#define KNUM   25     // 5x5 kernel taps
#define FIN    64
#define FOUT   64
#define XPITCH 66     // LDS row pitch (pad 64 -> 66, keeps 8B align, kills bank conflicts)

// ---------- kernel 1: zero bucket counters ----------
__global__ void k_zero(int* counts) {
  if (threadIdx.x < 32) counts[threadIdx.x] = 0;
}

// ---------- kernel 2: spline basis + bucket counting ----------
// degree=1, kernel=(5,5), open=(1,1):  v = coord*4; 4 taps (kx,ky) in {0,1}^2
__global__ void k_basis_count(const float* __restrict__ coord,
                              const float* __restrict__ mask, int E,
                              int* __restrict__ counts,
                              int* __restrict__ wiArr,
                              float* __restrict__ coeffArr) {
  int e = blockIdx.x * blockDim.x + threadIdx.x;
  if (e >= E) return;
  float vx = coord[2 * e] * 4.0f;
  float vy = coord[2 * e + 1] * 4.0f;
  float flx = floorf(vx), fly = floorf(vy);
  float fx = vx - flx, fy = vy - fly;
  int ix = (int)flx, iy = (int)fly;
  float msk = mask[e];
#pragma unroll
  for (int s = 0; s < 4; ++s) {
    int kx = s & 1, ky = (s >> 1) & 1;
    float bx = kx ? fx : 1.0f - fx;
    float by = ky ? fy : 1.0f - fy;
    float b = bx * by;
    int wx = (ix + kx) % 5; if (wx < 0) wx += 5;
    int wy = (iy + ky) % 5; if (wy < 0) wy += 5;
    int wi = wx + 5 * wy;
    if (!(b == b)) { b = 0.0f; wi = 0; }   // NaN guard, matches reference
    wiArr[4 * e + s]   = wi;
    coeffArr[4 * e + s] = b * msk;          // fold mask into coefficient
    atomicAdd(&counts[wi], 1);
  }
}

// ---------- kernel 3: tiny prefix sums (25 buckets) ----------
__global__ void k_scan(const int* __restrict__ counts,
                       int* __restrict__ offsets,
                       int* __restrict__ cursor,
                       int* __restrict__ tileOff) {
  if (threadIdx.x == 0 && blockIdx.x == 0) {
    int off = 0, t = 0;
    for (int k = 0; k < KNUM; ++k) {
      offsets[k] = off; cursor[k] = off; tileOff[k] = t;
      off += counts[k];
      t += (counts[k] + 15) >> 4;
    }
    offsets[KNUM] = off;
    tileOff[KNUM] = t;
  }
}

// ---------- kernel 4: scatter entries into buckets ----------
__global__ void k_scatter(const int* __restrict__ wiArr,
                          const float* __restrict__ coeffArr, int n,
                          int* __restrict__ cursor,
                          int* __restrict__ eidx,
                          float* __restrict__ coeffS) {
  int i = blockIdx.x * blockDim.x + threadIdx.x;
  if (i >= n) return;
  int wi = wiArr[i];
  int pos = atomicAdd(&cursor[wi], 1);
  eidx[pos]  = i >> 2;       // node index
  coeffS[pos] = coeffArr[i];
}

// ---------- kernel 5: out = mask * (x @ root + bias)  (fp32 WMMA) ----------
// 128 threads = 4 waves; block handles 16 nodes x 64 fout; wave -> 16 columns.
__global__ void __launch_bounds__(128)
k_root_gemm(const float* __restrict__ x,
            const float* __restrict__ mask,
            const float* __restrict__ root,
            const float* __restrict__ bias,
            float* __restrict__ out) {
  __shared__ float sX[16 * XPITCH];
  int tile = blockIdx.x;
  int tid = threadIdx.x;
  {
    int row = tid >> 3;
    int cb  = (tid & 7) * 8;
    const float* src = x + (size_t)(tile * 16 + row) * FIN + cb;
    float* dst = sX + row * XPITCH + cb;
#pragma unroll
    for (int j = 0; j < 8; ++j) dst[j] = src[j];
  }
  __syncthreads();

  int lane = tid & 31;
  int wave = tid >> 5;
  int m  = lane & 15;
  int kb = (lane >> 4) << 1;          // A: lanes 0-15 K{0,1}, lanes 16-31 K{2,3}
  int n  = wave * 16 + (lane & 15);
  v8f c = {};
#pragma unroll
  for (int k0 = 0; k0 < FIN; k0 += 4) {
    v2f a, b;
    a.x = sX[m * XPITCH + k0 + kb];
    a.y = sX[m * XPITCH + k0 + kb + 1];
    b.x = root[(k0 + kb) * FOUT + n];
    b.y = root[(k0 + kb + 1) * FOUT + n];
    c = __builtin_amdgcn_wmma_f32_16x16x4_f32(false, a, false, b,
                                              (short)0, c, false, false);
  }
  float bi = bias[n];
  int rbase = (lane >> 4) << 3;       // C/D: VGPR r -> rows r and r+8
#pragma unroll
  for (int r = 0; r < 8; ++r) {
    int node = tile * 16 + rbase + r;
    out[(size_t)node * FOUT + n] = (c[r] + bi) * mask[node];
  }
}

// ---------- kernel 6: per-bucket gathered GEMM + scatter-add ----------
// One block per 16-entry tile of a bucket. A rows = basis*mask-scaled x rows.
__global__ void __launch_bounds__(128)
k_bucket_gemm(const float* __restrict__ x,
              const float* __restrict__ weight,
              const int* __restrict__ offsets,
              const int* __restrict__ tileOff,
              const int* __restrict__ eidx,
              const float* __restrict__ coeffS,
              float* __restrict__ out) {
  int t = blockIdx.x;
  if (t >= tileOff[KNUM]) return;      // uniform across block
  int k = 0;
  while (t >= tileOff[k + 1]) ++k;     // 25-entry search, uniform
  int base = offsets[k] + (t - tileOff[k]) * 16;
  int cnt = offsets[k + 1] - base;
  if (cnt > 16) cnt = 16;

  __shared__ float sX[16 * XPITCH];
  __shared__ int sE[16];
  int tid = threadIdx.x;
  {
    int row = tid >> 3;
    int cb  = (tid & 7) * 8;
    float cf = 0.0f; int e = 0;
    if (row < cnt) { e = eidx[base + row]; cf = coeffS[base + row]; }
    if (tid < 16) sE[tid] = (tid < cnt) ? eidx[base + tid] : 0;
    const float* src = x + (size_t)e * FIN + cb;
    float* dst = sX + row * XPITCH + cb;
#pragma unroll
    for (int j = 0; j < 8; ++j) dst[j] = (row < cnt) ? cf * src[j] : 0.0f;
  }
  __syncthreads();

  const float* W = weight + (size_t)k * (FIN * FOUT);
  int lane = tid & 31;
  int wave = tid >> 5;
  int m  = lane & 15;
  int kb = (lane >> 4) << 1;
  int n  = wave * 16 + (lane & 15);
  v8f c = {};
#pragma unroll
  for (int k0 = 0; k0 < FIN; k0 += 4) {
    v2f a, b;
    a.x = sX[m * XPITCH + k0 + kb];
    a.y = sX[m * XPITCH + k0 + kb + 1];
    b.x = W[(k0 + kb) * FOUT + n];
    b.y = W[(k0 + kb + 1) * FOUT + n];
    c = __builtin_amdgcn_wmma_f32_16x16x4_f32(false, a, false, b,
                                              (short)0, c, false, false);
  }
  int rbase = (lane >> 4) << 3;
#pragma unroll
  for (int r = 0; r < 8; ++r) {
    int row = rbase + r;
    if (row < cnt)
      unsafeAtomicAdd(&out[(size_t)sE[row] * FOUT + n], c[r]);
  }
}

// ---------------------------------------------------------------------------
extern "C" void kernel_launch(void* const* d_in, const int* in_sizes, int n_in,
                              void* d_out, int out_size, void* d_ws, size_t ws_size,
                              hipStream_t stream) {
  const float* x      = (const float*)d_in[0];
  const float* coord  = (const float*)d_in[1];
  const float* mask   = (const float*)d_in[2];
  const float* weight = (const float*)d_in[3];
  const float* root   = (const float*)d_in[4];
  const float* bias   = (const float*)d_in[5];
  float* out = (float*)d_out;

  int E = in_sizes[2];               // mask has B*N elements

  // workspace layout (~2.1 MB for E=32768)
  int*   counts   = (int*)d_ws;
  int*   offsets  = counts + 32;
  int*   cursor   = offsets + 32;
  int*   tileOff  = cursor + 32;
  int*   wiArr    = tileOff + 32;                       // E*4 ints
  float* coeffArr = (float*)(wiArr + 4 * (size_t)E);    // E*4 floats
  int*   eidx     = (int*)(coeffArr + 4 * (size_t)E);   // E*4 ints
  float* coeffS   = (float*)(eidx + 4 * (size_t)E);     // E*4 floats

  k_zero<<<1, 32, 0, stream>>>(counts);
  k_basis_count<<<(E + 255) / 256, 256, 0, stream>>>(coord, mask, E, counts,
                                                     wiArr, coeffArr);
  k_scan<<<1, 32, 0, stream>>>(counts, offsets, cursor, tileOff);
  k_scatter<<<(4 * E + 255) / 256, 256, 0, stream>>>(wiArr, coeffArr, 4 * E,
                                                     cursor, eidx, coeffS);
  k_root_gemm<<<E / 16, 128, 0, stream>>>(x, mask, root, bias, out);

  int maxTiles = (4 * E) / 16 + KNUM;  // upper bound on per-bucket tiles
  k_bucket_gemm<<<maxTiles, 128, 0, stream>>>(x, weight, offsets, tileOff,
                                              eidx, coeffS, out);
}